// IntegratedLoss_27462020890713
// MI455X (gfx1250) — compile-verified
//
#include <hip/hip_runtime.h>
#include <math.h>

typedef __attribute__((ext_vector_type(2))) float v2f;
typedef __attribute__((ext_vector_type(8))) float v8f;

#define MD_THRES  0.5f
#define POS_MAX   1024
#define SL1_BETA  (1.0f/9.0f)
#define ALPHA_F   0.25f
#define GMAX_LDS  128     // per-block LDS pre-reduction slots for per-gt max

__device__ __forceinline__ float md_val(float s, float f) {
    // M_ALPHA=1, M_BETA=0, M_VAR=1  ->  |sa - |fa - sa||
    return fabsf(s - fabsf(f - s));
}

// ---------------- K1: md pass: per-anchor max/argmax over G, per-gt packed max over A
// LDS pre-reduction (ds_max_u64) cuts global atomics 8x; float2 loads halve VMEM ops.
__global__ void k1_md(const float* __restrict__ sa, const float* __restrict__ fa,
                      float* __restrict__ iou_max, int* __restrict__ iou_arg,
                      unsigned long long* __restrict__ gtpack, int A, int G) {
    __shared__ unsigned long long gmax[GMAX_LDS];
    int b = blockIdx.y;
    int a = blockIdx.x * blockDim.x + threadIdx.x;
    bool act = a < A;
    for (int g = threadIdx.x; g < G && g < GMAX_LDS; g += blockDim.x) gmax[g] = 0ull;
    __syncthreads();

    size_t row = ((size_t)b * A + (act ? a : 0)) * (size_t)G;
    const float* sp = sa + row;
    const float* fp = fa + row;
    float amax = -1.0f; int aarg = 0;
    unsigned int idx = act ? (unsigned int)a : 0xFFFFFFFFu;
    unsigned long long idxbits = (unsigned long long)(~idx);

    auto reduceg = [&](float m, int g) {
        unsigned long long pk =
            ((unsigned long long)__float_as_uint(m) << 32) | idxbits;
        // wave32 max-reduce (value, then lowest index via ~idx)
        for (int off = 16; off > 0; off >>= 1) {
            unsigned long long o = __shfl_down(pk, (unsigned)off, 32);
            if (o > pk) pk = o;
        }
        if ((threadIdx.x & 31) == 0) {
            if (g < GMAX_LDS) atomicMax(&gmax[g], pk);                  // ds_max_u64
            else atomicMax(&gtpack[(size_t)b * G + g], pk);             // rare fallback
        }
    };

    if ((G & 1) == 0) {
        // row base = ia*G*4 bytes; G even -> always 8B aligned -> float2 legal
        const float2* sp2 = reinterpret_cast<const float2*>(sp);
        const float2* fp2 = reinterpret_cast<const float2*>(fp);
        int H = G >> 1;
        for (int h = 0; h < H; ++h) {
            float m0 = 0.0f, m1 = 0.0f;
            if (act) {
                float2 sv = sp2[h];
                float2 fv = fp2[h];
                m0 = md_val(sv.x, fv.x);
                m1 = md_val(sv.y, fv.y);
                int g0 = 2 * h;
                if (m0 > amax) { amax = m0; aarg = g0; }
                if (m1 > amax) { amax = m1; aarg = g0 + 1; }
            }
            reduceg(m0, 2 * h);
            reduceg(m1, 2 * h + 1);
        }
    } else {
        for (int g = 0; g < G; ++g) {
            float m = 0.0f;
            if (act) {
                m = md_val(sp[g], fp[g]);
                if (m > amax) { amax = m; aarg = g; }
            }
            reduceg(m, g);
        }
    }

    if (act) {
        iou_max[(size_t)b * A + a] = amax;
        iou_arg[(size_t)b * A + a] = aarg;
    }
    __syncthreads();
    for (int g = threadIdx.x; g < G && g < GMAX_LDS; g += blockDim.x)
        atomicMax(&gtpack[(size_t)b * G + g], gmax[g]);
}

// ---------------- K2: forced anchors + gt counts
__global__ void k2_force(const float* __restrict__ ann,
                         const unsigned long long* __restrict__ gtpack,
                         int* __restrict__ forced, int* __restrict__ gt_count,
                         int A, int G, int B) {
    int i = blockIdx.x * blockDim.x + threadIdx.x;
    if (i >= B * G) return;
    int b = i / G;
    float label = ann[(size_t)i * 6 + 5];
    bool gt_ok = (label != -1.0f);
    if (gt_ok) atomicAdd(&gt_count[b], 1);
    unsigned long long pk = gtpack[i];
    float maxgt = __uint_as_float((unsigned int)(pk >> 32));
    unsigned int arg = ~((unsigned int)(pk & 0xFFFFFFFFull));
    if (gt_ok && (maxgt < MD_THRES) && arg < (unsigned int)A)
        atomicAdd(&forced[(size_t)b * A + arg], 1);
}

// ---------------- K3: per-batch prefix scan -> slots, pos_num
__global__ void k3_scan(const float* __restrict__ iou_max, const int* __restrict__ forced,
                        unsigned char* __restrict__ posflag, int* __restrict__ slotArr,
                        int* __restrict__ pos_num, int A) {
    __shared__ int lds[1024];
    int b = blockIdx.x;
    int tid = threadIdx.x;
    int running = 0;
    for (int base = 0; base < A; base += 1024) {
        int i = base + tid;
        int p = 0;
        if (i < A)
            p = ((iou_max[(size_t)b * A + i] >= MD_THRES) ||
                 (forced[(size_t)b * A + i] > 0)) ? 1 : 0;
        lds[tid] = p;
        __syncthreads();
        for (int off = 1; off < 1024; off <<= 1) {
            int t = (tid >= off) ? lds[tid - off] : 0;
            __syncthreads();
            lds[tid] += t;
            __syncthreads();
        }
        int incl = lds[tid];
        int total = lds[1023];
        if (i < A) {
            int rank = running + incl - p;
            bool valid = p && (rank < POS_MAX);
            posflag[(size_t)b * A + i] = (unsigned char)p;
            slotArr[(size_t)b * A + i] = valid ? rank : -1;
        }
        running += total;
        __syncthreads();
    }
    if (tid == 0) pos_num[b] = (running < POS_MAX) ? running : POS_MAX;
}

// ---------------- K4: compaction (recompute md rows for positives only)
__global__ void k4_compact(const float* __restrict__ sa, const float* __restrict__ fa,
                           const float* __restrict__ reg, const float* __restrict__ anchors,
                           const float* __restrict__ ann,
                           const int* __restrict__ slotArr, const int* __restrict__ iou_arg,
                           float* __restrict__ pos, float* __restrict__ pos_reg,
                           float* __restrict__ pos_rois, float* __restrict__ pos_gt,
                           int A, int G) {
    int b = blockIdx.y;
    int a = blockIdx.x * blockDim.x + threadIdx.x;
    if (a >= A) return;
    size_t ia = (size_t)b * A + a;
    int s = slotArr[ia];
    if (s < 0) return;
    const float* sp = sa + ia * (size_t)G;
    const float* fp = fa + ia * (size_t)G;
    float* pd = pos + ((size_t)b * POS_MAX + s) * (size_t)G;
    for (int g = 0; g < G; ++g) pd[g] = md_val(sp[g], fp[g]);
    const float* rp = reg + ia * 5;
    const float* ap = anchors + ia * 5;
    int ag = iou_arg[ia];
    const float* gp = ann + ((size_t)b * G + ag) * 6;
    float* prd = pos_reg  + ((size_t)b * POS_MAX + s) * 5;
    float* pad = pos_rois + ((size_t)b * POS_MAX + s) * 5;
    float* pgd = pos_gt   + ((size_t)b * POS_MAX + s) * 5;
    for (int i = 0; i < 5; ++i) { prd[i] = rp[i]; pad[i] = ap[i]; pgd[i] = gp[i]; }
}

// ---------------- K5: max/argmax over slots per (b,g)
__global__ void k5_maxpos(const float* __restrict__ pos,
                          float* __restrict__ max_pos, int* __restrict__ argmax_pos, int G) {
    __shared__ unsigned long long lds[256];
    int g = blockIdx.x, b = blockIdx.y;
    int tid = threadIdx.x;
    unsigned long long pk = 0;
    for (int s = tid; s < POS_MAX; s += 256) {
        float v = pos[((size_t)b * POS_MAX + s) * (size_t)G + g];
        unsigned long long c =
            ((unsigned long long)__float_as_uint(v) << 32) |
            (unsigned long long)(~((unsigned int)s));
        if (c > pk) pk = c;
    }
    lds[tid] = pk; __syncthreads();
    for (int off = 128; off > 0; off >>= 1) {
        if (tid < off && lds[tid + off] > lds[tid]) lds[tid] = lds[tid + off];
        __syncthreads();
    }
    if (tid == 0) {
        unsigned long long r = lds[0];
        max_pos[(size_t)b * G + g] = __uint_as_float((unsigned int)(r >> 32));
        argmax_pos[(size_t)b * G + g] = (int)(~((unsigned int)(r & 0xFFFFFFFFull)));
    }
}

// ---------------- K6: match_w per (b,slot)
__global__ void k6_matchw(const float* __restrict__ pos, const float* __restrict__ max_pos,
                          const int* __restrict__ argmax_pos, const int* __restrict__ gt_count,
                          float* __restrict__ match_w, int G) {
    int b = blockIdx.y;
    int s = blockIdx.x * blockDim.x + threadIdx.x;
    if (s >= POS_MAX) return;
    int nv = gt_count[b];
    float mw = -1e30f;
    const float* pr = pos + ((size_t)b * POS_MAX + s) * (size_t)G;
    for (int g = 0; g < G; ++g) {
        float pv = pr[g];
        bool mask = (pv >= MD_THRES) ||
                    ((g < nv) && (argmax_pos[(size_t)b * G + g] == s));
        float val = mask ? (1.0f - max_pos[(size_t)b * G + g]) + pv : pv + pv;
        mw = fmaxf(mw, val);
    }
    match_w[(size_t)b * POS_MAX + s] = mw;
}

// ---------------- K7: focal classification loss, block partial sums
__global__ void k7_cls(const float* __restrict__ cls, const float* __restrict__ iou_max_a,
                       const int* __restrict__ iou_arg, const unsigned char* __restrict__ posflag,
                       const int* __restrict__ slotArr, const float* __restrict__ ann,
                       const float* __restrict__ match_w, float* __restrict__ cls_part,
                       int A, int G, int C, int ppad) {
    __shared__ float lds[256];
    int b = blockIdx.y;
    int a = blockIdx.x * 256 + threadIdx.x;
    float sum = 0.0f;
    if (a < A) {
        size_t ia = (size_t)b * A + a;
        float im = iou_max_a[ia];
        bool positive = posflag[ia] != 0;
        int s = slotArr[ia];
        bool valid = s >= 0;
        float mw = 0.0f;
        if (valid) mw = match_w[(size_t)b * POS_MAX + s] + 1.0f;
        int ag = iou_arg[ia];
        int lbl = (int)ann[((size_t)b * G + ag) * 6 + 5];
        lbl = ((lbl % C) + C) % C;                 // python-style mod (handles -1)
        const float* cp = cls + ia * (size_t)C;
        for (int c = 0; c < C; ++c) {
            float p = fminf(fmaxf(cp[c], 1e-4f), 1.0f - 1e-4f);
            bool onehot = (c == lbl);
            float t = -1.0f;
            if ((im < MD_THRES - 0.1f) || positive) t = 0.0f;
            if (positive && onehot) t = 1.0f;
            float soft = (t == 0.0f) ? 1.0f : 0.0f;
            if (valid && onehot) soft = mw;
            if (t != -1.0f) {
                float af = (t == 1.0f) ? ALPHA_F : (1.0f - ALPHA_F);
                float q  = (t == 1.0f) ? (1.0f - p) : p;
                float fw = af * q * q;             // GAMMA = 2
                float bce = -(t * logf(p + 1e-6f) + (1.0f - t) * logf(1.0f - p + 1e-6f));
                sum += fw * bce * soft;
            }
        }
    }
    lds[threadIdx.x] = sum; __syncthreads();
    for (int off = 128; off > 0; off >>= 1) {
        if (threadIdx.x < off) lds[threadIdx.x] += lds[threadIdx.x + off];
        __syncthreads();
    }
    if (threadIdx.x == 0) cls_part[(size_t)b * ppad + blockIdx.x] = lds[0];
}

// ---------------- K8: smooth-L1 regression loss, one block per batch
__global__ void k8_reg(const float* __restrict__ pos_reg, const float* __restrict__ pos_rois,
                       const float* __restrict__ pos_gt, const float* __restrict__ match_w,
                       float* __restrict__ reg_sum) {
    __shared__ float lds[1024];
    int b = blockIdx.x, s = threadIdx.x;
    size_t base = ((size_t)b * POS_MAX + s) * 5;
    const float eps = 1e-6f;
    float ax = pos_rois[base+0], ay = pos_rois[base+1], aw = pos_rois[base+2],
          ah = pos_rois[base+3], at = pos_rois[base+4];
    float gx = pos_gt[base+0], gy = pos_gt[base+1], gw = pos_gt[base+2],
          gh = pos_gt[base+3], gth = pos_gt[base+4];
    aw = fmaxf(aw, eps); ah = fmaxf(ah, eps);
    gw = fmaxf(gw, eps); gh = fmaxf(gh, eps);
    float tgt[5];
    tgt[0] = (gx - ax) / aw;
    tgt[1] = (gy - ay) / ah;
    tgt[2] = logf(gw / aw);
    tgt[3] = logf(gh / ah);
    tgt[4] = (gth - at) * 0.017453292519943295f;
    float mw = match_w[(size_t)b * POS_MAX + s];
    float sum = 0.0f;
    for (int i = 0; i < 5; ++i) {
        float d = fabsf(pos_reg[base + i] - tgt[i]);
        float l = (d < SL1_BETA) ? 0.5f * d * d / SL1_BETA : d - 0.5f * SL1_BETA;
        sum += l * mw;
    }
    lds[s] = sum; __syncthreads();
    for (int off = 512; off > 0; off >>= 1) {
        if (s < off) lds[s] += lds[s + off];
        __syncthreads();
    }
    if (s == 0) reg_sum[b] = lds[0];
}

// ---------------- K9: WMMA-accelerated final reduction (one wave per batch)
// v_wmma_f32_16x16x4_f32 with an all-ones B: D[m][n] = sum_k A[m][k] + C
// -> row sums accumulate in C across chunks of 64 partials.
__global__ void k9_final(const float* __restrict__ cls_part, const float* __restrict__ reg_sum,
                         const int* __restrict__ pos_num, float* __restrict__ out,
                         int B, int ppad) {
    __shared__ float csum[32];
    int wave = threadIdx.x >> 5;
    int lane = threadIdx.x & 31;
    int b = wave;
    v2f bones; bones.x = 1.0f; bones.y = 1.0f;   // all-ones 4x16 B matrix
    v8f c = {0.f,0.f,0.f,0.f,0.f,0.f,0.f,0.f};
    int m  = lane & 15;              // A-matrix row (ISA 7.12.2: 32-bit A 16x4)
    int k0 = (lane >> 4) << 1;       // lanes 0-15 hold K=0,1; lanes 16-31 hold K=2,3
    const float* p = cls_part + (size_t)b * ppad;
    for (int base = 0; base < ppad; base += 64) {
        v2f av;
        av.x = p[base + m * 4 + k0];
        av.y = p[base + m * 4 + k0 + 1];
        c = __builtin_amdgcn_wmma_f32_16x16x4_f32(false, av, false, bones,
                                                  (short)0, c, false, false);
    }
    // D layout: VGPR v, lanes 0-15 -> row M=v; lanes 16-31 -> row M=v+8 (any N column)
    float s = c[0]+c[1]+c[2]+c[3]+c[4]+c[5]+c[6]+c[7];
    s += __shfl_down(s, 16u, 32);    // lane0: rows 0-7 + rows 8-15 = full sum
    if (lane == 0) csum[b] = s;
    __syncthreads();
    if (threadIdx.x == 0) {
        float lc = 0.0f, lr = 0.0f;
        for (int i = 0; i < B; ++i) {
            float pn = (float)pos_num[i];
            lc += csum[i] / fmaxf(pn, 1.0f);
            lr += reg_sum[i] / (pn * 5.0f);
        }
        out[0] = lc / (float)B;
        out[1] = lr / (float)B;
    }
}

extern "C" void kernel_launch(void* const* d_in, const int* in_sizes, int n_in,
                              void* d_out, int out_size, void* d_ws, size_t ws_size,
                              hipStream_t stream) {
    const float* cls     = (const float*)d_in[0];  // (B,A,C)
    const float* reg     = (const float*)d_in[1];  // (B,A,5)
    const float* anchors = (const float*)d_in[2];  // (B,A,5)
    const float* sa      = (const float*)d_in[3];  // (B,A,G)
    const float* fa      = (const float*)d_in[4];  // (B,A,G)
    const float* ann     = (const float*)d_in[5];  // (B,G,6)
    float* out = (float*)d_out;

    long long BA = (long long)in_sizes[1] / 5;
    int C = (int)((long long)in_sizes[0] / BA);
    int G = (int)((long long)in_sizes[3] / BA);
    int B = (int)((long long)in_sizes[5] / (6LL * G));
    int A = (int)(BA / B);

    // workspace carve-up (256B aligned)
    char* w = (char*)d_ws;
    size_t off = 0;
    auto alloc = [&](size_t bytes) -> void* {
        void* ptr = (void*)(w + off);
        off = (off + bytes + 255) & ~(size_t)255;
        return ptr;
    };
    unsigned long long* gtpack = (unsigned long long*)alloc(8ull * B * G);
    float* iou_max   = (float*)alloc(4ull * B * A);
    int*   iou_arg   = (int*)  alloc(4ull * B * A);
    int*   forced    = (int*)  alloc(4ull * B * A);
    int*   gt_count  = (int*)  alloc(4ull * B);
    unsigned char* posflag = (unsigned char*)alloc((size_t)B * A);
    int*   slotArr   = (int*)  alloc(4ull * B * A);
    int*   pos_num   = (int*)  alloc(4ull * B);
    float* posbuf    = (float*)alloc(4ull * B * POS_MAX * G);
    float* pos_reg   = (float*)alloc(4ull * B * POS_MAX * 5);
    float* pos_rois  = (float*)alloc(4ull * B * POS_MAX * 5);
    float* pos_gt    = (float*)alloc(4ull * B * POS_MAX * 5);
    float* max_pos   = (float*)alloc(4ull * B * G);
    int*   argmax_pos= (int*)  alloc(4ull * B * G);
    float* match_w   = (float*)alloc(4ull * B * POS_MAX);
    int nblkA = (A + 255) / 256;
    int ppad  = ((nblkA + 63) / 64) * 64;
    float* cls_part  = (float*)alloc(4ull * B * ppad);
    float* reg_sum   = (float*)alloc(4ull * B);
    (void)ws_size; (void)n_in; (void)out_size;

    hipMemsetAsync(d_ws, 0, off, stream);   // zero all scratch (capturable)

    dim3 gA((unsigned)nblkA, (unsigned)B);
    k1_md<<<gA, 256, 0, stream>>>(sa, fa, iou_max, iou_arg, gtpack, A, G);
    k2_force<<<(B * G + 127) / 128, 128, 0, stream>>>(ann, gtpack, forced, gt_count, A, G, B);
    k3_scan<<<B, 1024, 0, stream>>>(iou_max, forced, posflag, slotArr, pos_num, A);
    k4_compact<<<gA, 256, 0, stream>>>(sa, fa, reg, anchors, ann, slotArr, iou_arg,
                                       posbuf, pos_reg, pos_rois, pos_gt, A, G);
    k5_maxpos<<<dim3((unsigned)G, (unsigned)B), 256, 0, stream>>>(posbuf, max_pos, argmax_pos, G);
    k6_matchw<<<dim3(POS_MAX / 256, (unsigned)B), 256, 0, stream>>>(posbuf, max_pos, argmax_pos,
                                                                    gt_count, match_w, G);
    k7_cls<<<gA, 256, 0, stream>>>(cls, iou_max, iou_arg, posflag, slotArr, ann,
                                   match_w, cls_part, A, G, C, ppad);
    k8_reg<<<B, 1024, 0, stream>>>(pos_reg, pos_rois, pos_gt, match_w, reg_sum);
    k9_final<<<1, 32 * B, 0, stream>>>(cls_part, reg_sum, pos_num, out, B, ppad);
}